// LieOTAlign_56985626084109
// MI455X (gfx1250) — compile-verified
//
#include <hip/hip_runtime.h>
#include <hip/hip_bf16.h>
#include <math.h>

// Problem constants (match reference setup_inputs: N=8, L=2048)
#define NB    8
#define LL    2048
#define NLTOT (NB * LL)
#define TILES (LL / 16)
#define EPS_R (0.5f)      // 1/EPSILON, EPSILON=2.0
#define BIGV  (1e4f)
#define NITER 10

typedef float v2f __attribute__((ext_vector_type(2)));
typedef float v8f __attribute__((ext_vector_type(8)));

// ---------------- workspace layout (floats) ----------------
// [0,      4NL)   packP: [N][L][4] = (|p|^2, f, mask_res, 0)
// [4NL,    8NL)   packT: [N][L][4] = (|q|^2, g, mask_res, 0)
// [8NL,   12NL)   pred4: [N][L][4] = (x,y,z,0)
// [12NL,  16NL)   true4: [N][L][4] = (x,y,z,0)
// then: uint mcount[NB]  (per-batch mask_res popcount)

__device__ __forceinline__ v8f tile_mm(v2f a, v2f b) {
  v8f c = {0.f, 0.f, 0.f, 0.f, 0.f, 0.f, 0.f, 0.f};
  return __builtin_amdgcn_wmma_f32_16x16x4_f32(false, a, false, b, (short)0, c,
                                               false, false);
}

__global__ void k_zero(unsigned int* mcount) {
  if (threadIdx.x < NB) mcount[threadIdx.x] = 0u;
}

__global__ void k_init(const float* __restrict__ pred,
                       const float* __restrict__ truep,
                       const unsigned char* __restrict__ mres,
                       float* __restrict__ ws, unsigned int* __restrict__ mcount) {
  int i = blockIdx.x * blockDim.x + threadIdx.x;
  if (i >= NLTOT) return;
  float px = pred[i * 3 + 0], py = pred[i * 3 + 1], pz = pred[i * 3 + 2];
  float tx = truep[i * 3 + 0], ty = truep[i * 3 + 1], tz = truep[i * 3 + 2];
  float mk = mres[i] ? 1.f : 0.f;
  float* packP = ws;
  float* packT = ws + 4 * NLTOT;
  float* pred4 = ws + 8 * NLTOT;
  float* true4 = ws + 12 * NLTOT;
  packP[i * 4 + 0] = px * px + py * py + pz * pz;
  packP[i * 4 + 1] = 0.f;  // f
  packP[i * 4 + 2] = mk;
  packP[i * 4 + 3] = 0.f;
  packT[i * 4 + 0] = tx * tx + ty * ty + tz * tz;
  packT[i * 4 + 1] = 0.f;  // g
  packT[i * 4 + 2] = mk;
  packT[i * 4 + 3] = 0.f;
  pred4[i * 4 + 0] = px; pred4[i * 4 + 1] = py;
  pred4[i * 4 + 2] = pz; pred4[i * 4 + 3] = 0.f;
  true4[i * 4 + 0] = tx; true4[i * 4 + 1] = ty;
  true4[i * 4 + 2] = tz; true4[i * 4 + 3] = 0.f;
  if (mk > 0.f) atomicAdd(&mcount[i / LL], 1u);
}

// Unified Sinkhorn half-sweep:
//   pot_i[i] = -log(cnt) - logsumexp_j( pot_j[j] - M[i,j] )
// where M[i,j] = (|a_i|^2 + |b_j|^2 - 2 a_i.b_j)/EPS - 0.05*[i==j], BIG if masked.
// f-sweep: i = pred rows (co_i=pred4, pk_i=packP), j = true cols.
// g-sweep: operands swapped (M is used transposed; formula is symmetric).
__global__ void k_sweep(const float* __restrict__ co_i, float* __restrict__ pk_i,
                        const float* __restrict__ co_j, const float* __restrict__ pk_j,
                        const unsigned int* __restrict__ mcount) {
  int wg = blockIdx.x;
  int n = wg / TILES, i0 = (wg % TILES) * 16;
  int lane = threadIdx.x & 31, p = lane & 15, hi = lane >> 4;
  size_t nb = (size_t)n * LL;

  // A operand + own-row data (loop invariant)
  float4 av = *(const float4*)(co_i + (nb + i0 + p) * 4);
  v2f a = {hi ? av.z : av.x, hi ? av.w : av.y};
  float a2r[8], m1r[8], dneg[8];
  float mx[8], sm[8];
#pragma unroll
  for (int r = 0; r < 8; ++r) {
    float4 rv = *(const float4*)(pk_i + (nb + i0 + hi * 8 + r) * 4);
    a2r[r] = rv.x;
    m1r[r] = rv.z;
    dneg[r] = (p == hi * 8 + r) ? -0.05f : 0.f;  // in-tile diagonal pattern
    mx[r] = -1e30f;
    sm[r] = 0.f;
  }

  for (int j0 = 0; j0 < LL; j0 += 16) {
    const float* cj = co_j + (nb + j0 + p) * 4;
    const float* pj = pk_j + (nb + j0 + p) * 4;
    __builtin_prefetch(cj + 64, 0, 3);  // next tile (global_prefetch_b8)
    __builtin_prefetch(pj + 64, 0, 3);
    float4 bv = *(const float4*)cj;
    v2f b = {hi ? bv.z : bv.x, hi ? bv.w : bv.y};
    v8f c = tile_mm(a, b);
    float4 jv = *(const float4*)pj;       // (|b|^2, pot, mask, 0)
    float db = (j0 == i0) ? 1.f : 0.f;    // only the aligned tile hits the diag
#pragma unroll
    for (int r = 0; r < 8; ++r) {
      float Mv = (a2r[r] + jv.x - 2.f * c[r]) * EPS_R;
      Mv = fmaf(db, dneg[r], Mv);
      Mv = (m1r[r] * jv.z > 0.f) ? Mv : BIGV;
      float v = jv.y - Mv;
      float nm = fmaxf(mx[r], v);
      sm[r] = sm[r] * __expf(mx[r] - nm) + __expf(v - nm);
      mx[r] = nm;
    }
  }
  // merge across the 16 lanes of each half
#pragma unroll
  for (int off = 1; off < 16; off <<= 1) {
#pragma unroll
    for (int r = 0; r < 8; ++r) {
      float om = __shfl_xor(mx[r], off, 32);
      float os = __shfl_xor(sm[r], off, 32);
      float nm = fmaxf(mx[r], om);
      sm[r] = sm[r] * __expf(mx[r] - nm) + os * __expf(om - nm);
      mx[r] = nm;
    }
  }
  if (p == 0) {
    float logc = -__logf(fmaxf((float)mcount[n], 1.f));
#pragma unroll
    for (int r = 0; r < 8; ++r)
      pk_i[(nb + i0 + hi * 8 + r) * 4 + 1] = logc - (mx[r] + __logf(sm[r]));
  }
}

// pi = exp(f + g - M) (nan_to_num) -> out; mapped = pi @ true -> out
__global__ void k_pi(const float* __restrict__ ws,
                     float* __restrict__ out_map, float* __restrict__ out_pi) {
  int wg = blockIdx.x;
  int n = wg / TILES, l0 = (wg % TILES) * 16;
  int lane = threadIdx.x & 31, p = lane & 15, hi = lane >> 4;
  size_t nb = (size_t)n * LL;
  const float* packP = ws;
  const float* packT = ws + 4 * NLTOT;
  const float* pred4 = ws + 8 * NLTOT;
  const float* true4 = ws + 12 * NLTOT;

  float4 av = *(const float4*)(pred4 + (nb + l0 + p) * 4);
  v2f a = {hi ? av.z : av.x, hi ? av.w : av.y};
  float p2r[8], fr[8], m1r[8], dneg[8];
  int lrow[8];
  float ax[8], ay[8], az[8];
#pragma unroll
  for (int r = 0; r < 8; ++r) {
    int l = l0 + hi * 8 + r;
    lrow[r] = l;
    float4 rv = *(const float4*)(packP + (nb + l) * 4);
    p2r[r] = rv.x; fr[r] = rv.y; m1r[r] = rv.z;
    dneg[r] = (p == hi * 8 + r) ? -0.05f : 0.f;
    ax[r] = 0.f; ay[r] = 0.f; az[r] = 0.f;
  }

  for (int m0 = 0; m0 < LL; m0 += 16) {
    const float* cj = true4 + (nb + m0 + p) * 4;
    const float* pj = packT + (nb + m0 + p) * 4;
    __builtin_prefetch(cj + 64, 0, 3);
    __builtin_prefetch(pj + 64, 0, 3);
    float4 tv = *(const float4*)cj;       // (tx,ty,tz,0)
    v2f b = {hi ? tv.z : tv.x, hi ? tv.w : tv.y};
    v8f c = tile_mm(a, b);
    float4 jv = *(const float4*)pj;       // (|q|^2, g, mask, 0)
    float db = (m0 == l0) ? 1.f : 0.f;
#pragma unroll
    for (int r = 0; r < 8; ++r) {
      float Mv = (p2r[r] + jv.x - 2.f * c[r]) * EPS_R;
      Mv = fmaf(db, dneg[r], Mv);
      Mv = (m1r[r] * jv.z > 0.f) ? Mv : BIGV;
      float pv = __expf(fr[r] + jv.y - Mv);
      if (!(pv < 3.0e38f)) pv = 0.f;  // nan_to_num (pv >= 0 by construction)
      out_pi[(nb + lrow[r]) * LL + m0 + p] = pv;
      ax[r] = fmaf(pv, tv.x, ax[r]);
      ay[r] = fmaf(pv, tv.y, ay[r]);
      az[r] = fmaf(pv, tv.z, az[r]);
    }
  }
#pragma unroll
  for (int off = 1; off < 16; off <<= 1) {
#pragma unroll
    for (int r = 0; r < 8; ++r) {
      ax[r] += __shfl_xor(ax[r], off, 32);
      ay[r] += __shfl_xor(ay[r], off, 32);
      az[r] += __shfl_xor(az[r], off, 32);
    }
  }
  if (p == 0) {
#pragma unroll
    for (int r = 0; r < 8; ++r) {
      size_t o = (nb + lrow[r]) * 3;
      out_map[o + 0] = ax[r]; out_map[o + 1] = ay[r]; out_map[o + 2] = az[r];
    }
  }
}

// Centroids -> H -> 3x3 SVD (Jacobi on H^T H) -> R -> TM-score
__global__ void k_stats(const float* __restrict__ pred,
                        const unsigned char* __restrict__ mres,
                        const float* __restrict__ mgen,
                        const float* __restrict__ mapd,
                        float* __restrict__ out_tm, float d0) {
  __shared__ float sd[256 * 16];
  __shared__ float Rc[15];  // R[9], cp[3], cq[3]
  int n = blockIdx.x, tid = threadIdx.x;
  float acc[16];
#pragma unroll
  for (int k = 0; k < 16; ++k) acc[k] = 0.f;

  for (int l = tid; l < LL; l += 256) {
    size_t i = (size_t)n * LL + l;
    float w = mres[i] ? 1.f : 0.f;
    float px = pred[i * 3 + 0], py = pred[i * 3 + 1], pz = pred[i * 3 + 2];
    float qx = mapd[i * 3 + 0], qy = mapd[i * 3 + 1], qz = mapd[i * 3 + 2];
    acc[0] += w * px; acc[1] += w * py; acc[2] += w * pz;
    acc[3] += w * qx; acc[4] += w * qy; acc[5] += w * qz;
    acc[6] += w * px * qx; acc[7]  += w * px * qy; acc[8]  += w * px * qz;
    acc[9] += w * py * qx; acc[10] += w * py * qy; acc[11] += w * py * qz;
    acc[12] += w * pz * qx; acc[13] += w * pz * qy; acc[14] += w * pz * qz;
    acc[15] += w;
  }
#pragma unroll
  for (int k = 0; k < 16; ++k) sd[tid * 16 + k] = acc[k];
  __syncthreads();
  for (int s = 128; s > 0; s >>= 1) {
    if (tid < s)
#pragma unroll
      for (int k = 0; k < 16; ++k) sd[tid * 16 + k] += sd[(tid + s) * 16 + k];
    __syncthreads();
  }
  if (tid == 0) {
    float W = fmaxf(sd[15], 1e-6f);
    float cp[3], cq[3];
    for (int k = 0; k < 3; ++k) { cp[k] = sd[k] / W; cq[k] = sd[3 + k] / W; }
    float H[9];
    for (int i2 = 0; i2 < 3; ++i2)
      for (int j = 0; j < 3; ++j)
        H[i2 * 3 + j] = sd[6 + i2 * 3 + j] - sd[i2] * sd[3 + j] / W;
    float fn = 0.f;
    for (int k = 0; k < 9; ++k) fn += H[k] * H[k];
    fn = fmaxf(sqrtf(fn), 1e-8f);
    float A[9];
    for (int k = 0; k < 9; ++k) A[k] = H[k] / fn;
    A[0] += 1e-4f; A[4] += 1e-4f; A[8] += 1e-4f;
    // B = A^T A, Jacobi eigendecomposition B = V diag V^T
    float Bm[9];
    for (int i2 = 0; i2 < 3; ++i2)
      for (int j = 0; j < 3; ++j)
        Bm[i2 * 3 + j] = A[0 * 3 + i2] * A[0 * 3 + j] +
                         A[1 * 3 + i2] * A[1 * 3 + j] +
                         A[2 * 3 + i2] * A[2 * 3 + j];
    float V[9] = {1.f, 0.f, 0.f, 0.f, 1.f, 0.f, 0.f, 0.f, 1.f};
    const int pi_[3] = {0, 0, 1}, qi_[3] = {1, 2, 2};
    for (int sw = 0; sw < 10; ++sw) {
      for (int t = 0; t < 3; ++t) {
        int pp = pi_[t], qq = qi_[t];
        float apq = Bm[pp * 3 + qq];
        if (fabsf(apq) < 1e-18f) continue;
        float tau = (Bm[qq * 3 + qq] - Bm[pp * 3 + pp]) / (2.f * apq);
        float tt = copysignf(1.f, tau) / (fabsf(tau) + sqrtf(1.f + tau * tau));
        float cc = 1.f / sqrtf(1.f + tt * tt), ss = tt * cc;
        for (int k = 0; k < 3; ++k) {
          float bkp = Bm[k * 3 + pp], bkq = Bm[k * 3 + qq];
          Bm[k * 3 + pp] = cc * bkp - ss * bkq;
          Bm[k * 3 + qq] = ss * bkp + cc * bkq;
        }
        for (int k = 0; k < 3; ++k) {
          float bpk = Bm[pp * 3 + k], bqk = Bm[qq * 3 + k];
          Bm[pp * 3 + k] = cc * bpk - ss * bqk;
          Bm[qq * 3 + k] = ss * bpk + cc * bqk;
        }
        for (int k = 0; k < 3; ++k) {
          float vkp = V[k * 3 + pp], vkq = V[k * 3 + qq];
          V[k * 3 + pp] = cc * vkp - ss * vkq;
          V[k * 3 + qq] = ss * vkp + cc * vkq;
        }
      }
    }
    float lam[3] = {Bm[0], Bm[4], Bm[8]};
    for (int i2 = 0; i2 < 2; ++i2)
      for (int j = i2 + 1; j < 3; ++j)
        if (lam[j] > lam[i2]) {
          float tl = lam[i2]; lam[i2] = lam[j]; lam[j] = tl;
          for (int k = 0; k < 3; ++k) {
            float tv = V[k * 3 + i2]; V[k * 3 + i2] = V[k * 3 + j]; V[k * 3 + j] = tv;
          }
        }
    // U via A*v0, A*v1 Gram-Schmidt; u2 = u0 x u1 (det(U)=+1)
    float Av0[3], Av1[3], u0[3], u1[3], u2[3];
    for (int r = 0; r < 3; ++r) {
      Av0[r] = A[r * 3 + 0] * V[0] + A[r * 3 + 1] * V[3] + A[r * 3 + 2] * V[6];
      Av1[r] = A[r * 3 + 0] * V[1] + A[r * 3 + 1] * V[4] + A[r * 3 + 2] * V[7];
    }
    float n0 = sqrtf(Av0[0] * Av0[0] + Av0[1] * Av0[1] + Av0[2] * Av0[2]) + 1e-20f;
    for (int r = 0; r < 3; ++r) u0[r] = Av0[r] / n0;
    float dp = u0[0] * Av1[0] + u0[1] * Av1[1] + u0[2] * Av1[2];
    float w1[3];
    for (int r = 0; r < 3; ++r) w1[r] = Av1[r] - dp * u0[r];
    float n1 = sqrtf(w1[0] * w1[0] + w1[1] * w1[1] + w1[2] * w1[2]) + 1e-20f;
    for (int r = 0; r < 3; ++r) u1[r] = w1[r] / n1;
    u2[0] = u0[1] * u1[2] - u0[2] * u1[1];
    u2[1] = u0[2] * u1[0] - u0[0] * u1[2];
    u2[2] = u0[0] * u1[1] - u0[1] * u1[0];
    float detV = V[0] * (V[4] * V[8] - V[5] * V[7]) -
                 V[1] * (V[3] * V[8] - V[5] * V[6]) +
                 V[2] * (V[3] * V[7] - V[4] * V[6]);
    for (int a2 = 0; a2 < 3; ++a2)
      for (int b2 = 0; b2 < 3; ++b2)
        Rc[a2 * 3 + b2] = u0[a2] * V[b2 * 3 + 0] + u1[a2] * V[b2 * 3 + 1] +
                          detV * u2[a2] * V[b2 * 3 + 2];
    for (int k = 0; k < 3; ++k) { Rc[9 + k] = cp[k]; Rc[12 + k] = cq[k]; }
  }
  __syncthreads();

  float R0 = Rc[0], R1 = Rc[1], R2 = Rc[2], R3 = Rc[3], R4 = Rc[4];
  float R5 = Rc[5], R6 = Rc[6], R7 = Rc[7], R8 = Rc[8];
  float cpx = Rc[9], cpy = Rc[10], cpz = Rc[11];
  float cqx = Rc[12], cqy = Rc[13], cqz = Rc[14];
  float d0sq = d0 * d0;
  float tacc = 0.f, macc = 0.f;
  for (int l = tid; l < LL; l += 256) {
    size_t i = (size_t)n * LL + l;
    float mg = mgen[i];
    float dx = pred[i * 3 + 0] - cpx, dy = pred[i * 3 + 1] - cpy, dz = pred[i * 3 + 2] - cpz;
    float pax = dx * R0 + dy * R3 + dz * R6 + cqx;
    float pay = dx * R1 + dy * R4 + dz * R7 + cqy;
    float paz = dx * R2 + dy * R5 + dz * R8 + cqz;
    float ex = pax - mapd[i * 3 + 0], ey = pay - mapd[i * 3 + 1], ez = paz - mapd[i * 3 + 2];
    float dsq = ex * ex + ey * ey + ez * ez;
    tacc += mg / (1.f + dsq / d0sq);
    macc += mg;
  }
  sd[tid * 16 + 0] = tacc;
  sd[tid * 16 + 1] = macc;
  __syncthreads();
  for (int s = 128; s > 0; s >>= 1) {
    if (tid < s) {
      sd[tid * 16 + 0] += sd[(tid + s) * 16 + 0];
      sd[tid * 16 + 1] += sd[(tid + s) * 16 + 1];
    }
    __syncthreads();
  }
  if (tid == 0) out_tm[n] = sd[0] / fmaxf(sd[1], 1e-6f);
}

extern "C" void kernel_launch(void* const* d_in, const int* in_sizes, int n_in,
                              void* d_out, int out_size, void* d_ws, size_t ws_size,
                              hipStream_t stream) {
  const float* pred  = (const float*)d_in[0];
  const float* truep = (const float*)d_in[1];
  const float* mgen  = (const float*)d_in[2];
  const unsigned char* mres = (const unsigned char*)d_in[3];  // jnp.bool_ = 1 byte

  float* ws = (float*)d_ws;
  float* packP = ws;
  float* packT = ws + 4 * NLTOT;
  float* pred4 = ws + 8 * NLTOT;
  float* true4 = ws + 12 * NLTOT;
  unsigned int* mcount = (unsigned int*)(ws + 16 * NLTOT);

  float* out_tm  = (float*)d_out;
  float* out_map = out_tm + NB;
  float* out_pi  = out_map + (size_t)NB * LL * 3;

  k_zero<<<1, 32, 0, stream>>>(mcount);
  k_init<<<NLTOT / 256, 256, 0, stream>>>(pred, truep, mres, ws, mcount);
  for (int it = 0; it < NITER; ++it) {
    // f-sweep: rows = pred side; reads g (packT), writes f (packP)
    k_sweep<<<NB * TILES, 32, 0, stream>>>(pred4, packP, true4, packT, mcount);
    // g-sweep: rows = true side; reads f (packP), writes g (packT)
    k_sweep<<<NB * TILES, 32, 0, stream>>>(true4, packT, pred4, packP, mcount);
  }
  k_pi<<<NB * TILES, 32, 0, stream>>>(ws, out_map, out_pi);

  double d0d = 1.24 * cbrt((double)((LL > 16 ? LL : 16) - 15)) - 1.8;
  if (d0d < 0.5) d0d = 0.5;
  k_stats<<<NB, 256, 0, stream>>>(pred, mres, mgen, out_map, out_tm, (float)d0d);
}